// ImportanceSampling_62534723829963
// MI455X (gfx1250) — compile-verified
//
#include <hip/hip_runtime.h>

typedef __attribute__((ext_vector_type(2))) float v2f;
typedef __attribute__((ext_vector_type(4))) float v4f;
typedef __attribute__((ext_vector_type(8))) float v8f;

#define LDIM  5
#define CDIM  256
#define HWDIM 32768     // 128*256 spatial positions
#define NPB   128       // positions per block (8 waves * 16)
#define NPW   16        // positions per wave (WMMA M dimension)

// ---------------------------------------------------------------------------
// Pass 1: per-position mask.
// Each wave32 owns 16 positions. Lane (h = lane>>4, m = lane&15) accumulates
// f32 partial dot products for position n = base+m over c = {4*ci + 2h + j}.
// The 4 partials per (n, k) sit exactly in the V_WMMA_F32_16X16X4_F32
// A-matrix slots (lanes 0-15 -> K0/K1, lanes 16-31 -> K2/K3), so one WMMA
// against an all-ones B reduces them for all 16 rows simultaneously.
// D layout: lane 0 holds rows 0..7 (one per VGPR), lane 16 holds rows 8..15.
// ---------------------------------------------------------------------------
__global__ __launch_bounds__(256) void is_mask_kernel(
    const float* __restrict__ nf,       // (L, C, HW)
    const float* __restrict__ mlp_w,    // (1, C)
    const float* __restrict__ mlp_b,    // (1,)
    float*       __restrict__ mask)     // (HW,)
{
  __shared__ float lds_w[CDIM];
  __shared__ float lds_s[8][4][NPW];
  __shared__ float lds_p[8][4][NPW];

  const int tid = threadIdx.x;
  lds_w[tid] = mlp_w[tid];              // blockDim.x == CDIM == 256
  __syncthreads();

  const int wave = tid >> 5;
  const int lane = tid & 31;
  const int m    = lane & 15;
  const int h    = lane >> 4;           // which lane-half -> which K pair
  const int n    = blockIdx.x * NPB + wave * NPW + m;

  float As[4][2] = {{0.f,0.f},{0.f,0.f},{0.f,0.f},{0.f,0.f}};  // ego . nbr_k
  float Ap[4][2] = {{0.f,0.f},{0.f,0.f},{0.f,0.f},{0.f,0.f}};  // w   . nbr_k

  #pragma unroll 2
  for (int ci = 0; ci < CDIM / 4; ++ci) {
    const int    c0  = ci * 4 + h * 2;
    const size_t idx = (size_t)c0 * HWDIM + (size_t)n;
    const float  e0  = nf[idx];                 // ego, layer 0
    const float  e1  = nf[idx + HWDIM];
    const float  w0  = lds_w[c0];
    const float  w1  = lds_w[c0 + 1];
    #pragma unroll
    for (int k = 0; k < 4; ++k) {
      const size_t b  = (size_t)(k + 1) * CDIM * HWDIM + idx;
      const float  x0 = nf[b];
      const float  x1 = nf[b + HWDIM];
      As[k][0] = fmaf(e0, x0, As[k][0]);
      As[k][1] = fmaf(e1, x1, As[k][1]);
      Ap[k][0] = fmaf(w0, x0, Ap[k][0]);
      Ap[k][1] = fmaf(w1, x1, Ap[k][1]);
    }
  }

  v2f ones; ones[0] = 1.f; ones[1] = 1.f;
  v8f cz = {};
  #pragma unroll
  for (int k = 0; k < 4; ++k) {
    v2f a;  a[0]  = As[k][0]; a[1]  = As[k][1];
    v2f ap; ap[0] = Ap[k][0]; ap[1] = Ap[k][1];
    // D[m, j] = sum_K A[m, K] * 1  -> row-reduction of the 4 partial slots
    v8f ds = __builtin_amdgcn_wmma_f32_16x16x4_f32(
        false, a,  false, ones, (short)0, cz, false, false);
    v8f dp = __builtin_amdgcn_wmma_f32_16x16x4_f32(
        false, ap, false, ones, (short)0, cz, false, false);
    if (m == 0) {                       // lane 0 -> rows 0..7, lane 16 -> 8..15
      #pragma unroll
      for (int r = 0; r < 8; ++r) {
        lds_s[wave][k][h * 8 + r] = ds[r];
        lds_p[wave][k][h * 8 + r] = dp[r];
      }
    }
  }
  __syncthreads();

  if (tid < NPB) {                      // one thread per position
    const int w2 = tid >> 4, n2 = tid & 15;
    const float s0 = lds_s[w2][0][n2] * 0.0625f;   // / sqrt(256)
    const float s1 = lds_s[w2][1][n2] * 0.0625f;
    const float s2 = lds_s[w2][2][n2] * 0.0625f;
    const float s3 = lds_s[w2][3][n2] * 0.0625f;
    const float p0 = lds_p[w2][0][n2];
    const float p1 = lds_p[w2][1][n2];
    const float p2 = lds_p[w2][2][n2];
    const float p3 = lds_p[w2][3][n2];
    const float mx = fmaxf(fmaxf(s0, s1), fmaxf(s2, s3));
    const float e0 = __expf(s0 - mx), e1 = __expf(s1 - mx);
    const float e2 = __expf(s2 - mx), e3 = __expf(s3 - mx);
    const float se = e0 + e1 + e2 + e3;
    const float d  = (e0*p0 + e1*p1 + e2*p2 + e3*p3) / se + mlp_b[0];
    // sigmoid(relu(d)) > 0.5  <=>  d > 0
    __builtin_nontemporal_store((d > 0.f) ? 1.f : 0.f,
                                mask + blockIdx.x * NPB + tid);
  }
}

// ---------------------------------------------------------------------------
// Pass 2: out = x * mask (ego layer passes through). float4 over n.
// Input reads stay regular-temporal (L2-resident after pass 1); output
// stores go out non-temporal so they do not evict the input from L2.
// ---------------------------------------------------------------------------
__global__ __launch_bounds__(256) void is_apply_kernel(
    const float* __restrict__ nf,
    const float* __restrict__ mask,
    float*       __restrict__ out)
{
  const size_t i = ((size_t)blockIdx.x * 256 + threadIdx.x) * 4;
  const size_t l = i / ((size_t)CDIM * HWDIM);
  const int    n = (int)(i & (HWDIM - 1));
  v4f x = *(const v4f*)(nf + i);
  if (l != 0) {                          // wave-uniform (boundaries 128-aligned)
    const v4f mk = *(const v4f*)(mask + n);
    x *= mk;
  }
  __builtin_nontemporal_store(x, (v4f*)(out + i));
}

extern "C" void kernel_launch(void* const* d_in, const int* in_sizes, int n_in,
                              void* d_out, int out_size, void* d_ws, size_t ws_size,
                              hipStream_t stream) {
  const float* nf    = (const float*)d_in[0];   // (5,256,128,256) f32
  const float* mlp_w = (const float*)d_in[1];   // (1,256) f32
  const float* mlp_b = (const float*)d_in[2];   // (1,) f32
  float*       out   = (float*)d_out;
  float*       mask  = (float*)d_ws;            // 32768 floats = 128 KB

  is_mask_kernel<<<HWDIM / NPB, 256, 0, stream>>>(nf, mlp_w, mlp_b, mask);

  const int total  = LDIM * CDIM * HWDIM;       // 41,943,040
  const int blocks = total / 4 / 256;           // 40,960
  is_apply_kernel<<<blocks, 256, 0, stream>>>(nf, mask, out);
}